// GraphConvolution_78056735638031
// MI455X (gfx1250) — compile-verified
//
#include <hip/hip_runtime.h>
#include <math.h>

// Problem constants (from reference)
#define B_  32
#define V_  512
#define K_  16
#define D_  128
#define E_  32
#define NE_ 8192
#define BV  (B_ * V_)        // 16384
#define DE  (D_ + E_)        // 160

typedef __attribute__((ext_vector_type(2))) float v2f;
typedef __attribute__((ext_vector_type(8))) float v8f;

// Padded LDS strides (floats). Chosen so row*stride mod 64 banks is distinct
// for m = 0..15 and stays 8-byte aligned for b64 fragment loads.
#define XS1 164   // kernel1: 160-wide rows
#define XS2 260   // kernel2: 256-wide rows
#define SS2 132   // kernel2: 128-wide support rows

__device__ __forceinline__ float gelu_exact(float x) {
    return 0.5f * x * (1.0f + erff(x * 0.70710678118654752440f));
}

// ---------------------------------------------------------------------------
// Kernel 1: per (b,v) neighbor gather + [16,160] @ W1^T -> [16,128],
// exact GELU, masked sum over the 16 neighbors -> nl[bv, 128].
// One wave (32 threads) per (b,v). 40 K-steps x 8 N-tiles of wmma f32 16x16x4.
// ---------------------------------------------------------------------------
__global__ __launch_bounds__(32)
void gc_neighbor_gemm_kernel(const float* __restrict__ vfeat,      // [BV, D]
                             const int*   __restrict__ atom_adj,   // [BV*K]
                             const int*   __restrict__ bond_adj,   // [BV*K]
                             const float* __restrict__ edge_init,  // [B*NE, E]
                             const float* __restrict__ nbs_mask,   // [BV*K]
                             const float* __restrict__ W1,         // [D, D+E]
                             const float* __restrict__ b1,         // [D]
                             float*       __restrict__ nl_out)     // [BV, D]
{
    __shared__ __align__(16) float Xs[16 * XS1];

    const int lane = threadIdx.x;            // 0..31
    const int bv   = blockIdx.x;             // 0..BV-1
    const int ln   = lane & 15;              // N within 16-tile / M row for A
    const int kb   = (lane >> 4) << 1;       // 0 or 2: K-pair select (A/B frag)
    const int hi8  = (lane >> 4) << 3;       // 0 or 8: M-half select (C/D frag)

    // ---- gather neighbor rows into LDS: X[k][0:128]=vn, X[k][128:160]=en ----
    for (int k = 0; k < K_; ++k) {
        const int aidx = atom_adj[bv * K_ + k];      // row into [BV, D]
        const int bidx = bond_adj[bv * K_ + k];      // row into [B*NE, E]
        float4 tv = ((const float4*)(vfeat + (size_t)aidx * D_))[lane];
        *(float4*)&Xs[k * XS1 + lane * 4] = tv;
        if (lane < 8) {
            float4 te = ((const float4*)(edge_init + (size_t)bidx * E_))[lane];
            *(float4*)&Xs[k * XS1 + D_ + lane * 4] = te;
        }
    }
    __syncthreads();

    // ---- GEMM: [16 x 160] @ W1^T -> [16 x 128] in 8 accumulators ----
    v8f acc[8];
#pragma unroll
    for (int nt = 0; nt < 8; ++nt) acc[nt] = (v8f)0.0f;

    for (int kk = 0; kk < DE / 4; ++kk) {            // 40 K-steps of 4
        const int k0 = kk * 4 + kb;
        // A fragment: rows m=ln, columns k0,k0+1 (per-lane-half K pair)
        v2f a = *(const v2f*)&Xs[ln * XS1 + k0];
#pragma unroll
        for (int nt = 0; nt < 8; ++nt) {
            // B fragment: B[k][n] = W1[n][k]; contiguous k-pair in W1 row n
            v2f bf = *(const v2f*)&W1[(size_t)(nt * 16 + ln) * DE + k0];
            acc[nt] = __builtin_amdgcn_wmma_f32_16x16x4_f32(
                false, a, false, bf, (short)0, acc[nt], false, false);
        }
    }

    // ---- epilogue: bias, GELU, masked reduce over M (neighbors) ----
    float nbm[8];
#pragma unroll
    for (int r = 0; r < 8; ++r) nbm[r] = nbs_mask[bv * K_ + hi8 + r];

#pragma unroll
    for (int nt = 0; nt < 8; ++nt) {
        const int n   = nt * 16 + ln;
        const float bias = b1[n];
        float s = 0.0f;
#pragma unroll
        for (int r = 0; r < 8; ++r) {
            float x = acc[nt][r] + bias;
            s += gelu_exact(x) * nbm[r];
        }
        // combine M=0..7 half (lanes 0-15) with M=8..15 half (lanes 16-31)
        s += __shfl_xor(s, 16, 32);
        if (lane < 16) nl_out[(size_t)bv * D_ + n] = s;
    }
}

// ---------------------------------------------------------------------------
// Kernel 2: 16 rows of BV per wave.
//   X = [nl | vf]  [16,256];  hi = X @ W2^T + b2
//   support = (1-alpha)*hi + alpha*h0
//   out = theta*(support @ W3^T + b3) + (1-theta)*support + vf
// ---------------------------------------------------------------------------
__global__ __launch_bounds__(32)
void gc_update_gemm_kernel(const float* __restrict__ nl,      // [BV, D]
                           const float* __restrict__ vfeat,   // [BV, D]
                           const float* __restrict__ h0,      // [BV, D]
                           const float* __restrict__ W2,      // [D, 2D]
                           const float* __restrict__ b2,      // [D]
                           const float* __restrict__ W3,      // [D, D]
                           const float* __restrict__ b3,      // [D]
                           const float* __restrict__ lamda,   // [1]
                           const float* __restrict__ alpha,   // [1]
                           const int*   __restrict__ lval,    // [1]
                           float*       __restrict__ out)     // [BV, D]
{
    __shared__ __align__(16) float Xs[16 * XS2];
    __shared__ __align__(16) float Ss[16 * SS2];

    const int lane = threadIdx.x;
    const int mt   = blockIdx.x;             // 0..BV/16-1
    const int ln   = lane & 15;
    const int kb   = (lane >> 4) << 1;
    const int hi8  = (lane >> 4) << 3;

    const float theta = logf(lamda[0] / (float)lval[0] + 1.0f);
    const float al    = alpha[0];

    // ---- stage X = [nl | vf] into LDS ----
    for (int r = 0; r < 16; ++r) {
        const size_t g = (size_t)(mt * 16 + r);
        float4 t0 = ((const float4*)(nl    + g * D_))[lane];
        *(float4*)&Xs[r * XS2 + lane * 4] = t0;
        float4 t1 = ((const float4*)(vfeat + g * D_))[lane];
        *(float4*)&Xs[r * XS2 + D_ + lane * 4] = t1;
    }
    __syncthreads();

    // ---- GEMM1: [16 x 256] @ W2^T -> hi [16 x 128] ----
    v8f acc[8];
#pragma unroll
    for (int nt = 0; nt < 8; ++nt) acc[nt] = (v8f)0.0f;

    for (int kk = 0; kk < (2 * D_) / 4; ++kk) {      // 64 K-steps
        const int k0 = kk * 4 + kb;
        v2f a = *(const v2f*)&Xs[ln * XS2 + k0];
#pragma unroll
        for (int nt = 0; nt < 8; ++nt) {
            v2f bf = *(const v2f*)&W2[(size_t)(nt * 16 + ln) * (2 * D_) + k0];
            acc[nt] = __builtin_amdgcn_wmma_f32_16x16x4_f32(
                false, a, false, bf, (short)0, acc[nt], false, false);
        }
    }

    // ---- support = (1-alpha)*(hi + b2) + alpha*h0, stash into LDS ----
#pragma unroll
    for (int nt = 0; nt < 8; ++nt) {
        const int n = nt * 16 + ln;
        const float bias2 = b2[n];
#pragma unroll
        for (int r = 0; r < 8; ++r) {
            const int m = hi8 + r;
            const size_t g = (size_t)(mt * 16 + m);
            float hv  = acc[nt][r] + bias2;
            float sup = (1.0f - al) * hv + al * h0[g * D_ + n];
            Ss[m * SS2 + n] = sup;
        }
    }
    __syncthreads();

    // ---- GEMM2: support [16 x 128] @ W3^T -> [16 x 128] ----
    v8f acc2[8];
#pragma unroll
    for (int nt = 0; nt < 8; ++nt) acc2[nt] = (v8f)0.0f;

    for (int kk = 0; kk < D_ / 4; ++kk) {            // 32 K-steps
        const int k0 = kk * 4 + kb;
        v2f a = *(const v2f*)&Ss[ln * SS2 + k0];
#pragma unroll
        for (int nt = 0; nt < 8; ++nt) {
            v2f bf = *(const v2f*)&W3[(size_t)(nt * 16 + ln) * D_ + k0];
            acc2[nt] = __builtin_amdgcn_wmma_f32_16x16x4_f32(
                false, a, false, bf, (short)0, acc2[nt], false, false);
        }
    }

    // ---- out = theta*(Y + b3) + (1-theta)*support + vf ----
#pragma unroll
    for (int nt = 0; nt < 8; ++nt) {
        const int n = nt * 16 + ln;
        const float bias3 = b3[n];
#pragma unroll
        for (int r = 0; r < 8; ++r) {
            const int m = hi8 + r;
            const size_t g = (size_t)(mt * 16 + m);
            float y   = acc2[nt][r] + bias3;
            float sup = Ss[m * SS2 + n];
            float o   = theta * y + (1.0f - theta) * sup + vfeat[g * D_ + n];
            out[g * D_ + n] = o;
        }
    }
}

// ---------------------------------------------------------------------------
// Launch
// ---------------------------------------------------------------------------
extern "C" void kernel_launch(void* const* d_in, const int* in_sizes, int n_in,
                              void* d_out, int out_size, void* d_ws, size_t ws_size,
                              hipStream_t stream) {
    (void)in_sizes; (void)n_in; (void)out_size; (void)ws_size;

    const float* vfeat     = (const float*)d_in[0];   // [B,V,D]
    const int*   atom_adj  = (const int*)  d_in[1];   // [B*V*K]
    const int*   bond_adj  = (const int*)  d_in[2];   // [B*V*K]
    const float* h0        = (const float*)d_in[3];   // [B,V,D]
    const float* lamda     = (const float*)d_in[4];   // [1]
    const float* alpha     = (const float*)d_in[5];   // [1]
    const int*   lval      = (const int*)  d_in[6];   // [1]
    const float* edge_init = (const float*)d_in[7];   // [B,NE,E]
    /* d_in[8] vertex_mask: unused by reference */
    const float* nbs_mask  = (const float*)d_in[9];   // [B,V,K]
    const float* W1        = (const float*)d_in[10];  // [D, D+E]
    const float* b1        = (const float*)d_in[11];  // [D]
    const float* W2        = (const float*)d_in[12];  // [D, 2D]
    const float* b2        = (const float*)d_in[13];  // [D]
    const float* W3        = (const float*)d_in[14];  // [D, D]
    const float* b3        = (const float*)d_in[15];  // [D]

    float* nl_ws = (float*)d_ws;                      // [BV, D] = 8 MB
    float* out   = (float*)d_out;

    gc_neighbor_gemm_kernel<<<dim3(BV), dim3(32), 0, stream>>>(
        vfeat, atom_adj, bond_adj, edge_init, nbs_mask, W1, b1, nl_ws);

    gc_update_gemm_kernel<<<dim3(BV / 16), dim3(32), 0, stream>>>(
        nl_ws, vfeat, h0, W2, b2, W3, b3, lamda, alpha, lval, out);
}